// MLPBlock_75600014344434
// MI455X (gfx1250) — compile-verified
//
#include <hip/hip_runtime.h>
#include <math.h>

#define H   768
#define I2  1536
#define NE  128
#define NB  32
#define TOPK 4

typedef float v8f __attribute__((ext_vector_type(8)));
typedef float v2f __attribute__((ext_vector_type(2)));

// ---------------- workspace layout (in floats) ----------------
// [0, 24576)            t (rmsnormed activations, B x H)
// [24576, 24704)        counts (int, per expert)
// [24704, 28800)        tok_id (int, E x 32)
// [28800, 32896)        tok_w  (float, E x 32)

__global__ __launch_bounds__(256)
void init_kernel(const float* __restrict__ x, float* __restrict__ out,
                 int* __restrict__ counts) {
    int idx = blockIdx.x * 256 + threadIdx.x;
    if (idx < NB * H) out[idx] = x[idx];           // residual init
    if (idx < NE)     counts[idx] = 0;
}

__global__ __launch_bounds__(256)
void gate_kernel(const float* __restrict__ x, const float* __restrict__ norm_scale,
                 const float* __restrict__ gate_w,
                 float* __restrict__ t_ws, int* __restrict__ counts,
                 int* __restrict__ tok_id, float* __restrict__ tok_w) {
    __shared__ float st[H];
    __shared__ float sg[NE];
    __shared__ float red[256];
    const int b = blockIdx.x, tid = threadIdx.x;

    // rmsnorm
    float local = 0.f;
    for (int i = tid; i < H; i += 256) { float v = x[b * H + i]; st[i] = v; local += v * v; }
    red[tid] = local;
    __syncthreads();
    for (int s = 128; s > 0; s >>= 1) { if (tid < s) red[tid] += red[tid + s]; __syncthreads(); }
    const float rs = rsqrtf(red[0] * (1.0f / (float)H) + 1e-5f);
    for (int i = tid; i < H; i += 256) {
        float tv = st[i] * rs * norm_scale[i];
        st[i] = tv;
        t_ws[b * H + i] = tv;
    }
    __syncthreads();

    // gate logits: g[e] = t . gate_w[:, e]   (gate_w is H x E row-major)
    if (tid < NE) {
        float acc = 0.f;
        for (int h = 0; h < H; ++h) acc += st[h] * gate_w[h * NE + tid];
        sg[tid] = acc;
    }
    __syncthreads();

    // top-4 + softmax + append to expert lists
    if (tid == 0) {
        int   bi[TOPK]; float bv[TOPK];
        unsigned taken[4] = {0u, 0u, 0u, 0u};
        for (int k = 0; k < TOPK; ++k) {
            float best = -1e30f; int bidx = 0;
            for (int e = 0; e < NE; ++e) {
                if (taken[e >> 5] & (1u << (e & 31))) continue;
                if (sg[e] > best) { best = sg[e]; bidx = e; }
            }
            bi[k] = bidx; bv[k] = best;
            taken[bidx >> 5] |= 1u << (bidx & 31);
        }
        float m = bv[0], sum = 0.f, ew[TOPK];
        for (int k = 0; k < TOPK; ++k) { ew[k] = __expf(bv[k] - m); sum += ew[k]; }
        for (int k = 0; k < TOPK; ++k) {
            float wgt = ew[k] / sum;
            int e = bi[k];
            int slot = atomicAdd(&counts[e], 1);
            tok_id[e * NB + slot] = b;
            tok_w[e * NB + slot]  = wgt;
        }
    }
}

// One workgroup per expert. 8 waves (wave32). Streams the expert's weights
// exactly once per 16-token group; compute via V_WMMA_F32_16X16X4_F32.
// k-loops are software-pipelined (double-buffered A fragments) so each wave
// keeps ~12-24 weight loads in flight instead of serializing on one temp.
__global__ __launch_bounds__(256)
void moe_kernel(const float* __restrict__ t_ws, const int* __restrict__ counts,
                const int* __restrict__ tok_id, const float* __restrict__ tok_w,
                const float* __restrict__ mlp1_w, const float* __restrict__ mlp1_b,
                const float* __restrict__ mlp2_w, const float* __restrict__ mlp2_b,
                float* __restrict__ out) {
    // TU reused: first holds T (activations), then U (swiglu output).
    // Element (k, tok) stored at ((k>>1)*16 + tok)*2 + (k&1) so each lane's
    // B-fragment (K = kb, kb+1 at fixed tok) is one aligned 8-byte ds load.
    __shared__ float TU[H * 16];                 // 48 KB
    __shared__ float sb1[I2];                    // mlp1 bias, 6 KB
    __shared__ float sb2[H];                     // mlp2 bias, 3 KB
    __shared__ int   stokb[16];
    __shared__ float stokw[16];

    const int e = blockIdx.x;
    const int n = counts[e];
    if (n == 0) return;

    const int tid  = threadIdx.x;
    const int lane = tid & 31;
    const int wave = tid >> 5;                   // 0..7
    const int l15  = lane & 15;
    const int hh   = lane >> 4;                  // half-wave

    const float* W1 = mlp1_w + (size_t)e * I2 * H;
    const float* W2 = mlp2_w + (size_t)e * H * H;

    // stage biases into LDS once (keeps them out of VGPRs during k-loops)
    for (int i = tid; i < I2; i += 256) sb1[i] = mlp1_b[(size_t)e * I2 + i];
    for (int i = tid; i < H;  i += 256) sb2[i] = mlp2_b[(size_t)e * H + i];

    const v8f vzero = {0.f, 0.f, 0.f, 0.f, 0.f, 0.f, 0.f, 0.f};

    for (int base = 0; base < n; base += 16) {
        const int np = min(16, n - base);
        if (tid < 16) {
            stokb[tid] = (tid < np) ? tok_id[e * NB + base + tid] : 0;
            stokw[tid] = (tid < np) ? tok_w[e * NB + base + tid]  : 0.f;
        }
        __syncthreads();

        // stage T (pad missing token columns with zero)
        for (int idx = tid; idx < 16 * H; idx += 256) {
            int tok = idx / H;
            int kk  = idx - tok * H;             // contiguous per thread-stride
            float v = (tok < np) ? t_ws[stokb[tok] * H + kk] : 0.f;
            TU[((kk >> 1) * 16 + tok) * 2 + (kk & 1)] = v;
        }
        __syncthreads();

        // ---------------- MLP1: 96 tiles of 16 out-rows ----------------
        const float* a1p[12];
        #pragma unroll
        for (int i = 0; i < 12; ++i)
            a1p[i] = W1 + (size_t)((wave + 8 * i) * 16 + l15) * H + 2 * hh;

        v8f acc[12];
        #pragma unroll
        for (int i = 0; i < 12; ++i) acc[i] = vzero;

        {
            v2f aA[12], aB[12];
            #pragma unroll
            for (int i = 0; i < 12; ++i) aA[i] = *(const v2f*)(a1p[i]);      // k = 0
            #pragma unroll 1
            for (int k = 0; k < H; k += 8) {
                const int kb0  = k + 2 * hh;
                const int kb1  = k + 4 + 2 * hh;
                const int kpre = (k + 8 < H) ? (k + 8) : 0;  // clamped prefetch
                v2f b0 = *(const v2f*)&TU[((kb0 >> 1) * 16 + l15) * 2];
                #pragma unroll
                for (int i = 0; i < 12; ++i) aB[i] = *(const v2f*)(a1p[i] + k + 4);
                #pragma unroll
                for (int i = 0; i < 12; ++i)
                    acc[i] = __builtin_amdgcn_wmma_f32_16x16x4_f32(
                        false, aA[i], false, b0, (short)0, acc[i], false, false);
                v2f b1 = *(const v2f*)&TU[((kb1 >> 1) * 16 + l15) * 2];
                #pragma unroll
                for (int i = 0; i < 12; ++i) aA[i] = *(const v2f*)(a1p[i] + kpre);
                #pragma unroll
                for (int i = 0; i < 12; ++i)
                    acc[i] = __builtin_amdgcn_wmma_f32_16x16x4_f32(
                        false, aB[i], false, b1, (short)0, acc[i], false, false);
            }
        }
        __syncthreads();   // all waves done reading T from TU

        // bias + swiglu in registers, write U over TU
        #pragma unroll
        for (int i = 0; i < 12; ++i) {
            const int rb = (wave + 8 * i) * 16 + 8 * hh;  // rows rb..rb+7 = acc[i][0..7]
            #pragma unroll
            for (int j = 0; j < 4; ++j) {
                float xg = acc[i][2 * j]     + sb1[rb + 2 * j];       // even chan: glu
                float xl = acc[i][2 * j + 1] + sb1[rb + 2 * j + 1];   // odd chan: linear
                xg = fminf(xg, 7.f);
                xl = fminf(fmaxf(xl, -7.f), 7.f);
                float og = xg / (1.f + __expf(-1.702f * xg));
                float u  = og * (xl + 1.f);
                const int ch = (rb + 2 * j) >> 1;                     // 0..767
                TU[((ch >> 1) * 16 + l15) * 2 + (ch & 1)] = u;
            }
        }
        __syncthreads();

        // ---------------- MLP2: 48 tiles of 16 out-rows ----------------
        const float* a2p[6];
        #pragma unroll
        for (int i = 0; i < 6; ++i)
            a2p[i] = W2 + (size_t)((wave + 8 * i) * 16 + l15) * H + 2 * hh;

        v8f acc2[6];
        #pragma unroll
        for (int i = 0; i < 6; ++i) acc2[i] = vzero;

        {
            v2f aA[6], aB[6];
            #pragma unroll
            for (int i = 0; i < 6; ++i) aA[i] = *(const v2f*)(a2p[i]);       // k = 0
            #pragma unroll 1
            for (int k = 0; k < H; k += 8) {
                const int kb0  = k + 2 * hh;
                const int kb1  = k + 4 + 2 * hh;
                const int kpre = (k + 8 < H) ? (k + 8) : 0;
                v2f b0 = *(const v2f*)&TU[((kb0 >> 1) * 16 + l15) * 2];
                #pragma unroll
                for (int i = 0; i < 6; ++i) aB[i] = *(const v2f*)(a2p[i] + k + 4);
                #pragma unroll
                for (int i = 0; i < 6; ++i)
                    acc2[i] = __builtin_amdgcn_wmma_f32_16x16x4_f32(
                        false, aA[i], false, b0, (short)0, acc2[i], false, false);
                v2f b1 = *(const v2f*)&TU[((kb1 >> 1) * 16 + l15) * 2];
                #pragma unroll
                for (int i = 0; i < 6; ++i) aA[i] = *(const v2f*)(a2p[i] + kpre);
                #pragma unroll
                for (int i = 0; i < 6; ++i)
                    acc2[i] = __builtin_amdgcn_wmma_f32_16x16x4_f32(
                        false, aB[i], false, b1, (short)0, acc2[i], false, false);
            }
        }

        // gate-weighted scatter into residual
        const int tok = l15;
        if (tok < np) {
            const float wgt = stokw[tok];
            const int   bo  = stokb[tok] * H;
            #pragma unroll
            for (int i = 0; i < 6; ++i) {
                const int rb = (wave + 8 * i) * 16 + 8 * hh;
                #pragma unroll
                for (int v = 0; v < 8; ++v) {
                    float val = (acc2[i][v] + sb2[rb + v]) * wgt;
                    __hip_atomic_fetch_add(&out[bo + rb + v], val,
                                           __ATOMIC_RELAXED, __HIP_MEMORY_SCOPE_AGENT);
                }
            }
        }
        __syncthreads();   // before next token-group overwrites TU
    }
}

extern "C" void kernel_launch(void* const* d_in, const int* in_sizes, int n_in,
                              void* d_out, int out_size, void* d_ws, size_t ws_size,
                              hipStream_t stream) {
    const float* x          = (const float*)d_in[0];
    const float* norm_scale = (const float*)d_in[1];
    const float* gate_w     = (const float*)d_in[2];
    const float* mlp1_w     = (const float*)d_in[3];
    const float* mlp1_b     = (const float*)d_in[4];
    const float* mlp2_w     = (const float*)d_in[5];
    const float* mlp2_b     = (const float*)d_in[6];
    float* out = (float*)d_out;

    float* ws     = (float*)d_ws;
    float* t_ws   = ws;                            // 24576 floats
    int*   counts = (int*)(ws + 24576);            // 128 ints
    int*   tok_id = (int*)(ws + 24576 + 128);      // 4096 ints
    float* tok_w  = ws + 24576 + 128 + 4096;       // 4096 floats

    init_kernel<<<96, 256, 0, stream>>>(x, out, counts);
    gate_kernel<<<NB, 256, 0, stream>>>(x, norm_scale, gate_w, t_ws, counts, tok_id, tok_w);
    moe_kernel<<<NE, 256, 0, stream>>>(t_ws, counts, tok_id, tok_w,
                                       mlp1_w, mlp1_b, mlp2_w, mlp2_b, out);
}